// Network_77635828842942
// MI455X (gfx1250) — compile-verified
//
#include <hip/hip_runtime.h>
#include <math.h>

// ---------------- constants ----------------
#define BB   64
#define II   512
#define CC   32
#define NN   2048          // B*C
#define PNW  64            // PN
#define PLW  16            // PL
#define PRED 96

typedef __attribute__((ext_vector_type(2))) float v2f;
typedef __attribute__((ext_vector_type(8))) float v8f;

__device__ __forceinline__ float gelu_f(float x) {
  return 0.5f * x * (1.0f + erff(x * 0.70710678118654752f));
}
__device__ __forceinline__ float softplus_f(float x) {
  return (x > 20.0f) ? x : log1pf(expf(x));
}
__device__ __forceinline__ float wave_max(float v) {
  for (int o = 16; o > 0; o >>= 1) v = fmaxf(v, __shfl_xor(v, o, 32));
  return v;
}
__device__ __forceinline__ float wave_sum(float v) {
  for (int o = 16; o > 0; o >>= 1) v += __shfl_xor(v, o, 32);
  return v;
}

// ---------------- register-blocked WMMA f32 16x16x4 GEMM ----------------
// One wave owns a (16*MB) x (16*NB) macro-tile; MB A-frags and NB B-frags are
// loaded per K-step and MB*NB v_wmma_f32_16x16x4_f32 issued against them.
// Out[m][n] = scale * sum_k A[m][k] * B(k,n)  (+bias, +gelu)
// BLAYOUT 0: B operand stored N x K row-major (weights W[out][in])
// BLAYOUT 1: B operand stored K x N row-major (row stride = Nc)
// OUTMODE 0: Out[m*Nc+n] = act(scale*v + bias[n])
// OUTMODE 1: Out[n*M+m] += pvec[b*4+wsel] * scale * v    (transposed accumulate)
template <int BLAYOUT, int OUTMODE, int MB, int NB>
__global__ __launch_bounds__(32) void gemm16(
    const float* __restrict__ Abase, const float* __restrict__ Bbase,
    float* __restrict__ Obase, int M, int Nc, int K,
    long sAb, long sBb, long sOb,
    const float* __restrict__ bias, float scale, int act,
    const float* __restrict__ pvec, int wsel) {
  const int b = blockIdx.z;
  const float* A  = Abase + (long)b * sAb;
  const float* Bm = Bbase + (long)b * sBb;
  float* O        = Obase + (long)b * sOb;

  const int m0   = blockIdx.x * (16 * MB);
  const int n0   = blockIdx.y * (16 * NB);
  const int lane = threadIdx.x;
  const int half = lane >> 4;     // 0: K pair {k,k+1}, 1: {k+2,k+3}
  const int lr   = lane & 15;

  v8f acc[MB][NB] = {};

  const float* arow[MB];
#pragma unroll
  for (int mi = 0; mi < MB; ++mi) {
    arow[mi] = A + (long)(m0 + mi * 16 + lr) * K + 2 * half;
    __builtin_prefetch(arow[mi], 0, 1);
  }
  const float* brow[NB];
  if (BLAYOUT == 0) {
#pragma unroll
    for (int ni = 0; ni < NB; ++ni) {
      brow[ni] = Bm + (long)(n0 + ni * 16 + lr) * K + 2 * half;
      __builtin_prefetch(brow[ni], 0, 1);
    }
  }

  for (int k = 0; k < K; k += 4) {
    v2f a[MB], bf[NB];
#pragma unroll
    for (int mi = 0; mi < MB; ++mi) a[mi] = *(const v2f*)(arow[mi] + k);
#pragma unroll
    for (int ni = 0; ni < NB; ++ni) {
      if (BLAYOUT == 0) {
        bf[ni] = *(const v2f*)(brow[ni] + k);
      } else {
        int kk = k + 2 * half;
        bf[ni].x = Bm[(long)kk * Nc + n0 + ni * 16 + lr];
        bf[ni].y = Bm[(long)(kk + 1) * Nc + n0 + ni * 16 + lr];
      }
    }
#pragma unroll
    for (int mi = 0; mi < MB; ++mi)
#pragma unroll
      for (int ni = 0; ni < NB; ++ni)
        acc[mi][ni] = __builtin_amdgcn_wmma_f32_16x16x4_f32(
            false, a[mi], false, bf[ni], (short)0, acc[mi][ni], false, false);
  }

  float pw = 1.0f;
  if (OUTMODE == 1) pw = pvec[b * 4 + wsel];

#pragma unroll
  for (int mi = 0; mi < MB; ++mi) {
#pragma unroll
    for (int ni = 0; ni < NB; ++ni) {
      int nn = n0 + ni * 16 + lr;
      float bv = 0.f;
      if (OUTMODE == 0 && bias) bv = bias[nn];
#pragma unroll
      for (int r = 0; r < 8; ++r) {
        int mm = m0 + mi * 16 + r + 8 * half;
        float v = acc[mi][ni][r] * scale;
        if (OUTMODE == 0) {
          v += bv;
          if (act == 1) v = gelu_f(v);
          O[(long)mm * Nc + nn] = v;
        } else {
          O[(long)nn * M + mm] += pw * v;
        }
      }
    }
  }
}

// ---------------- small kernels ----------------

// per-batch: smean(C), h=gelu(smean@rw1^T+rb1), p=softmax(h@rw2^T+rb2)
__global__ __launch_bounds__(32) void head_kernel(
    const float* __restrict__ s_in, const float* __restrict__ rw1,
    const float* __restrict__ rb1, const float* __restrict__ rw2,
    const float* __restrict__ rb2, float* __restrict__ pout) {
  __shared__ float smv[CC];
  __shared__ float hv[16];
  int b = blockIdx.x, c = threadIdx.x;
  float acc = 0.f;
  for (int l = 0; l < II; ++l) acc += s_in[((long)b * II + l) * CC + c];
  smv[c] = acc * (1.0f / (float)II);
  __syncthreads();
  if (c < 16) {
    float a = rb1[c];
    for (int j = 0; j < CC; ++j) a += smv[j] * rw1[c * CC + j];
    hv[c] = gelu_f(a);
  }
  __syncthreads();
  if (c == 0) {
    float lo[3], mx = -1e30f;
    for (int w = 0; w < 3; ++w) {
      float a = rb2[w];
      for (int j = 0; j < 16; ++j) a += hv[j] * rw2[w * 16 + j];
      lo[w] = a; mx = fmaxf(mx, a);
    }
    float s = 0.f;
    for (int w = 0; w < 3; ++w) { lo[w] = expf(lo[w] - mx); s += lo[w]; }
    for (int w = 0; w < 3; ++w) pout[b * 4 + w] = lo[w] / s;
    pout[b * 4 + 3] = 0.f;
  }
}

// out[(b*C+c)*I + l] = in[(b*I+l)*C + c]   (used for s->s2 init and t->tmat)
__global__ __launch_bounds__(32) void transpose_bic(const float* __restrict__ in,
                                                    float* __restrict__ out) {
  int bl = blockIdx.x;             // b*I + l
  int b = bl >> 9, l = bl & 511, c = threadIdx.x;
  out[((long)(b * CC + c)) * II + l] = in[(long)bl * CC + c];
}

// phimean[b][l] = sum_w p[b][w] * mean_c Ptab_w[(cyc+l)%Wk][c]
__global__ void phim_kernel(const float* __restrict__ P0, const float* __restrict__ P1,
                            const float* __restrict__ P2, const int* __restrict__ cyc,
                            const float* __restrict__ p, float* __restrict__ phim) {
  int gid = blockIdx.x * blockDim.x + threadIdx.x;
  if (gid >= BB * II) return;
  int b = gid >> 9, l = gid & 511;
  const float* Pt[3] = {P0, P1, P2};
  const int Wk[3] = {24, 168, 336};
  float acc = 0.f;
  for (int w = 0; w < 3; ++w) {
    int idx = (cyc[b] + l) % Wk[w];
    const float* r = Pt[w] + idx * CC;
    float s = 0.f;
    for (int c = 0; c < CC; ++c) s += r[c];
    acc += p[b * 4 + w] * s * (1.0f / (float)CC);
  }
  phim[gid] = acc;
}

// Qr[(b*I+l)*C + c] = softplus(Atab[idx][c]) * cos(Ptab[idx][c])
__global__ __launch_bounds__(32) void qr_kernel(const float* __restrict__ Atab,
                                                const float* __restrict__ Ptab,
                                                const int* __restrict__ cyc, int Wk,
                                                float* __restrict__ Qr) {
  int bl = blockIdx.x;
  int b = bl >> 9, l = bl & 511, c = threadIdx.x;
  int idx = (cyc[b] + l) % Wk;
  Qr[(long)bl * CC + c] = softplus_f(Atab[idx * CC + c]) * cosf(Ptab[idx * CC + c]);
}

// in-place softmax over rows of length 512 (one wave per row)
__global__ __launch_bounds__(32) void softmax512(float* __restrict__ S) {
  float* row = S + (long)blockIdx.x * II;
  int lane = threadIdx.x;
  float v[16], mx = -1e30f;
  for (int i = 0; i < 16; ++i) { v[i] = row[lane + 32 * i]; mx = fmaxf(mx, v[i]); }
  mx = wave_max(mx);
  float sum = 0.f;
  for (int i = 0; i < 16; ++i) { v[i] = expf(v[i] - mx); sum += v[i]; }
  sum = wave_sum(sum);
  float r = 1.0f / sum;
  for (int i = 0; i < 16; ++i) row[lane + 32 * i] = v[i] * r;
}

__global__ void zero_stats(float* __restrict__ stats) {
  if (threadIdx.x < 256) stats[threadIdx.x] = 0.f;
}

// fused patch attention: windows, qkv, hd=2 attention with cos bias, Wo, gelu, BN partials
__global__ __launch_bounds__(64) void patch_attn(
    const float* __restrict__ s2, const float* __restrict__ phim,
    const float* __restrict__ Wqkv, const float* __restrict__ bqkv,
    const float* __restrict__ Wo, const float* __restrict__ bo,
    const float* __restrict__ alpha_p, float* __restrict__ sx,
    float* __restrict__ stats) {
  __shared__ float sp[PNW][PLW];
  __shared__ float qkv[PNW][48];
  __shared__ float bias[PNW][PNW];
  __shared__ float phis[PNW];
  __shared__ float ob[PNW][PLW];
  __shared__ float bsum[PNW], bsq[PNW];
  const int n = blockIdx.x, b = n >> 5, tid = threadIdx.x;
  const float alpha = alpha_p[0];

  for (int idx = tid; idx < PNW * PLW; idx += 64) {
    int pn = idx >> 4, pl = idx & 15;
    int col = pn * 8 + pl; if (col > II - 1) col = II - 1;
    sp[pn][pl] = s2[(long)n * II + col];
  }
  {
    int pn = tid; float a = 0.f;
    for (int pl = 0; pl < PLW; ++pl) {
      int col = pn * 8 + pl; if (col > II - 1) col = II - 1;
      a += phim[(long)b * II + col];
    }
    phis[pn] = a * (1.0f / (float)PLW);
    bsum[tid] = 0.f; bsq[tid] = 0.f;
  }
  __syncthreads();
  for (int idx = tid; idx < PNW * 48; idx += 64) {
    int pn = idx / 48, j = idx % 48;
    float a = bqkv[j];
    for (int m = 0; m < PLW; ++m) a += sp[pn][m] * Wqkv[j * PLW + m];
    qkv[pn][j] = a;
  }
  for (int idx = tid; idx < PNW * PNW; idx += 64) {
    int tt = idx >> 6, ss = idx & 63;
    bias[tt][ss] = alpha * cosf(phis[tt] - phis[ss]);
  }
  __syncthreads();
  {
    const int t = tid;
    const float isq2 = 0.70710678118654752f;  // 1/sqrt(hd), hd=2
    for (int h = 0; h < 8; ++h) {
      float q0 = qkv[t][2 * h], q1 = qkv[t][2 * h + 1];
      float mx = -1e30f;
      for (int s_ = 0; s_ < PNW; ++s_) {
        float sc = (q0 * qkv[s_][16 + 2 * h] + q1 * qkv[s_][17 + 2 * h]) * isq2 + bias[t][s_];
        mx = fmaxf(mx, sc);
      }
      float sum = 0.f, o0 = 0.f, o1 = 0.f;
      for (int s_ = 0; s_ < PNW; ++s_) {
        float sc = (q0 * qkv[s_][16 + 2 * h] + q1 * qkv[s_][17 + 2 * h]) * isq2 + bias[t][s_];
        float e = expf(sc - mx);
        sum += e;
        o0 += e * qkv[s_][32 + 2 * h];
        o1 += e * qkv[s_][33 + 2 * h];
      }
      ob[t][2 * h] = o0 / sum; ob[t][2 * h + 1] = o1 / sum;
    }
  }
  __syncthreads();
  for (int idx = tid; idx < PNW * PLW; idx += 64) {
    int pn = idx >> 4, pl = idx & 15;
    float a = bo[pl];
    for (int m = 0; m < PLW; ++m) a += ob[pn][m] * Wo[pl * PLW + m];
    float val = gelu_f(sp[pn][pl] + a);
    sx[(long)n * (PNW * PLW) + idx] = val;
    atomicAdd(&bsum[pn], val);
    atomicAdd(&bsq[pn], val * val);
  }
  __syncthreads();
  atomicAdd(&stats[tid], bsum[tid]);
  atomicAdd(&stats[64 + tid], bsq[tid]);
}

__global__ __launch_bounds__(64) void bn_final(float* __restrict__ stats) {
  int pn = threadIdx.x;
  const float cnt = (float)(NN * PLW);
  float mu = stats[pn] / cnt;
  float var = stats[64 + pn] / cnt - mu * mu;
  stats[128 + pn] = mu;
  stats[192 + pn] = rsqrtf(var + 1e-5f);
}

__global__ void bn_apply(const float* __restrict__ sx, const float* __restrict__ stats,
                         const float* __restrict__ g, const float* __restrict__ bb,
                         float* __restrict__ sxn) {
  long gid = (long)blockIdx.x * blockDim.x + threadIdx.x;
  if (gid >= (long)NN * PNW * PLW) return;
  int pn = ((int)(gid & 1023)) >> 4;
  sxn[gid] = (sx[gid] - stats[128 + pn]) * stats[192 + pn] * g[pn] + bb[pn];
}

// pair-mean then layernorm: in (n,Lin) -> out (n,Lout), Lout=Lin/2, Lout<=192
__global__ __launch_bounds__(64) void pair_ln(const float* __restrict__ in,
                                              float* __restrict__ out,
                                              const float* __restrict__ g,
                                              const float* __restrict__ bb,
                                              int Lin, int Lout) {
  __shared__ float row[192];
  __shared__ float red[64], red2[64];
  int n = blockIdx.x, tid = threadIdx.x;
  float ls = 0.f, lq = 0.f;
  for (int j = tid; j < Lout; j += 64) {
    float v = 0.5f * (in[(long)n * Lin + 2 * j] + in[(long)n * Lin + 2 * j + 1]);
    row[j] = v; ls += v; lq += v * v;
  }
  red[tid] = ls; red2[tid] = lq;
  __syncthreads();
  for (int st = 32; st > 0; st >>= 1) {
    if (tid < st) { red[tid] += red[tid + st]; red2[tid] += red2[tid + st]; }
    __syncthreads();
  }
  float m = red[0] / (float)Lout;
  float var = red2[0] / (float)Lout - m * m;
  float inv = rsqrtf(var + 1e-5f);
  for (int j = tid; j < Lout; j += 64)
    out[(long)n * Lout + j] = (row[j] - m) * inv * g[j] + bb[j];
}

// u += irfft(rfft(u)*mask(<5)) == circular conv with lowpass kernel
__global__ __launch_bounds__(96) void fftlp_kernel(const float* __restrict__ u4,
                                                   float* __restrict__ uf) {
  __shared__ float hf[PRED];
  __shared__ float row[PRED];
  int n = blockIdx.x, tid = threadIdx.x;
  const float w1 = 6.28318530717958648f / (float)PRED;
  float d = (float)tid;
  hf[tid] = (1.0f + 2.0f * (cosf(w1 * d) + cosf(2.f * w1 * d) +
                            cosf(3.f * w1 * d) + cosf(4.f * w1 * d))) * (1.0f / (float)PRED);
  row[tid] = u4[(long)n * PRED + tid];
  __syncthreads();
  float acc = row[tid];
  for (int j = 0; j < PRED; ++j) {
    int dd = tid - j; if (dd < 0) dd += PRED;
    acc += row[j] * hf[dd];
  }
  uf[(long)n * PRED + tid] = acc;
}

// 3-tap channel-mixing conv over length-96 axis + residual gelu
__global__ __launch_bounds__(256) void conv_kernel(const float* __restrict__ uf,
                                                   const float* __restrict__ te_w,
                                                   const float* __restrict__ te_b,
                                                   float* __restrict__ uo) {
  __shared__ float ut[CC][PRED];
  int b = blockIdx.x, tid = threadIdx.x;
  for (int idx = tid; idx < CC * PRED; idx += 256) {
    int c = idx / PRED, k = idx % PRED;
    ut[c][k] = uf[((long)(b * CC + c)) * PRED + k];
  }
  __syncthreads();
  for (int idx = tid; idx < CC * PRED; idx += 256) {
    int co = idx / PRED, k = idx % PRED;
    float acc = te_b[co];
    for (int ci = 0; ci < CC; ++ci) {
      const float* w = te_w + (co * CC + ci) * 3;
      float l_ = (k > 0) ? ut[ci][k - 1] : 0.f;
      float m_ = ut[ci][k];
      float r_ = (k < PRED - 1) ? ut[ci][k + 1] : 0.f;
      acc += l_ * w[0] + m_ * w[1] + r_ * w[2];
    }
    uo[((long)(b * CC + co)) * PRED + k] = ut[co][k] + gelu_f(acc);
  }
}

__global__ void cat_kernel(const float* __restrict__ sxo, const float* __restrict__ uo,
                           float* __restrict__ cat) {
  long gid = (long)blockIdx.x * blockDim.x + threadIdx.x;
  if (gid >= (long)NN * 192) return;
  int n = (int)(gid / 192), j = (int)(gid % 192);
  cat[gid] = (j < 96) ? sxo[(long)n * 96 + j] : uo[(long)n * 96 + (j - 96)];
}

// out[b][p][c] = x[(b*C+c)*96 + p]
__global__ void out_kernel(const float* __restrict__ x, float* __restrict__ out) {
  long gid = (long)blockIdx.x * blockDim.x + threadIdx.x;
  if (gid >= (long)BB * PRED * CC) return;
  int c = (int)(gid & 31);
  int p = (int)((gid >> 5) % PRED);
  int b = (int)(gid / (PRED * CC));
  out[gid] = x[((long)(b * CC + c)) * PRED + p];
}

// ---------------- host driver ----------------
extern "C" void kernel_launch(void* const* d_in, const int* in_sizes, int n_in,
                              void* d_out, int out_size, void* d_ws, size_t ws_size,
                              hipStream_t stream) {
  (void)in_sizes; (void)n_in; (void)out_size; (void)ws_size;
  const float* s_in  = (const float*)d_in[0];
  const float* t_in  = (const float*)d_in[1];
  const int*   cyc   = (const int*)d_in[2];
  const float* rw1   = (const float*)d_in[3];
  const float* rb1   = (const float*)d_in[4];
  const float* rw2   = (const float*)d_in[5];
  const float* rb2   = (const float*)d_in[6];
  const float* A0    = (const float*)d_in[7];
  const float* A1    = (const float*)d_in[8];
  const float* A2    = (const float*)d_in[9];
  const float* P0    = (const float*)d_in[10];
  const float* P1    = (const float*)d_in[11];
  const float* P2    = (const float*)d_in[12];
  const float* Wqkv  = (const float*)d_in[13];
  const float* bqkv  = (const float*)d_in[14];
  const float* Wo    = (const float*)d_in[15];
  const float* bo    = (const float*)d_in[16];
  const float* alpha = (const float*)d_in[17];
  const float* bn_g  = (const float*)d_in[18];
  const float* bn_b  = (const float*)d_in[19];
  const float* W3    = (const float*)d_in[20];
  const float* b3    = (const float*)d_in[21];
  const float* W4    = (const float*)d_in[22];
  const float* b4    = (const float*)d_in[23];
  const float* W5    = (const float*)d_in[24];
  const float* b5    = (const float*)d_in[25];
  const float* W6    = (const float*)d_in[26];
  const float* b6    = (const float*)d_in[27];
  const float* W7    = (const float*)d_in[28];
  const float* b7    = (const float*)d_in[29];
  const float* W8    = (const float*)d_in[30];
  const float* b8    = (const float*)d_in[31];
  const float* ln1g  = (const float*)d_in[32];
  const float* ln1b  = (const float*)d_in[33];
  const float* ln2g  = (const float*)d_in[34];
  const float* ln2b  = (const float*)d_in[35];
  const float* te_w  = (const float*)d_in[36];
  const float* te_b  = (const float*)d_in[37];

  float* ws = (float*)d_ws;
  // region A (reused): wavelet-phase S, then post-phase intermediates
  float* S    = ws + 0;                 // 64*512*512
  float* sx   = ws + 0;                 // 2048*1024
  float* sxn  = ws + 2097152;           // 2048*1024
  float* h3   = ws + 4194304;           // 2048*192
  float* sxo  = ws + 4587520;           // 2048*96
  float* tmat = ws + 4784128;           // 2048*512
  float* u5   = ws + 5832704;           // 2048*384
  float* u1   = ws + 6619136;           // 2048*192
  float* u2   = ws + 7012352;           // 2048*96
  float* u3   = ws + 7208960;           // 2048*48
  float* u4   = ws + 7307264;           // 2048*96
  float* uf   = ws + 7503872;           // 2048*96
  float* uo   = ws + 7700480;           // 2048*96
  float* cat  = ws + 7897088;           // 2048*192
  float* xbuf = ws + 8290304;           // 2048*96
  // persistent regions
  float* Qr    = ws + 16777216;         // 64*512*32
  float* s2    = ws + 17825792;         // 64*32*512
  float* phim  = ws + 18874368;         // 64*512
  float* pvec  = ws + 18907136;         // 64*4
  float* stats = ws + 18907392;         // 256

  head_kernel<<<BB, 32, 0, stream>>>(s_in, rw1, rb1, rw2, rb2, pvec);
  transpose_bic<<<BB * II, 32, 0, stream>>>(s_in, s2);
  phim_kernel<<<(BB * II + 255) / 256, 256, 0, stream>>>(P0, P1, P2, cyc, pvec, phim);

  const float* Atabs[3] = {A0, A1, A2};
  const float* Ptabs[3] = {P0, P1, P2};
  const int    Wks[3]   = {24, 168, 336};
  for (int w = 0; w < 3; ++w) {
    qr_kernel<<<BB * II, 32, 0, stream>>>(Atabs[w], Ptabs[w], cyc, Wks[w], Qr);
    // S[b] = (1/sqrt(C)) * Qr[b](512x32) @ s_in[b](512x32)^T   -- 64x64 macro-tiles
    gemm16<0, 0, 4, 4><<<dim3(8, 8, BB), 32, 0, stream>>>(
        Qr, s_in, S, II, II, CC, 16384, 16384, 262144,
        nullptr, 0.17677669529663689f, 0, nullptr, 0);
    softmax512<<<BB * II, 32, 0, stream>>>(S);
    // s2[b] (CxI) += p[b][w] * (Aw[b](512x512) @ s_in[b](512x32))^T  -- 64x32 tiles
    gemm16<1, 1, 4, 2><<<dim3(8, 1, BB), 32, 0, stream>>>(
        S, s_in, s2, II, CC, II, 262144, 16384, 16384,
        nullptr, 1.0f, 0, pvec, w);
  }

  zero_stats<<<1, 256, 0, stream>>>(stats);
  patch_attn<<<NN, 64, 0, stream>>>(s2, phim, Wqkv, bqkv, Wo, bo, alpha, sx, stats);
  bn_final<<<1, 64, 0, stream>>>(stats);
  bn_apply<<<(NN * 1024) / 256, 256, 0, stream>>>(sx, stats, bn_g, bn_b, sxn);

  gemm16<0, 0, 4, 4><<<dim3(32, 3, 1), 32, 0, stream>>>(
      sxn, W3, h3, NN, 192, 1024, 0, 0, 0, b3, 1.0f, 1, nullptr, 0);
  gemm16<0, 0, 4, 2><<<dim3(32, 3, 1), 32, 0, stream>>>(
      h3, W4, sxo, NN, 96, 192, 0, 0, 0, b4, 1.0f, 0, nullptr, 0);

  transpose_bic<<<BB * II, 32, 0, stream>>>(t_in, tmat);
  gemm16<0, 0, 4, 4><<<dim3(32, 6, 1), 32, 0, stream>>>(
      tmat, W5, u5, NN, 384, 512, 0, 0, 0, b5, 1.0f, 0, nullptr, 0);
  pair_ln<<<NN, 64, 0, stream>>>(u5, u1, ln1g, ln1b, 384, 192);
  gemm16<0, 0, 4, 2><<<dim3(32, 3, 1), 32, 0, stream>>>(
      u1, W6, u2, NN, 96, 192, 0, 0, 0, b6, 1.0f, 0, nullptr, 0);
  pair_ln<<<NN, 64, 0, stream>>>(u2, u3, ln2g, ln2b, 96, 48);
  gemm16<0, 0, 4, 2><<<dim3(32, 3, 1), 32, 0, stream>>>(
      u3, W7, u4, NN, 96, 48, 0, 0, 0, b7, 1.0f, 0, nullptr, 0);

  fftlp_kernel<<<NN, 96, 0, stream>>>(u4, uf);
  conv_kernel<<<BB, 256, 0, stream>>>(uf, te_w, te_b, uo);

  cat_kernel<<<(NN * 192 + 255) / 256, 256, 0, stream>>>(sxo, uo, cat);
  gemm16<0, 0, 4, 2><<<dim3(32, 3, 1), 32, 0, stream>>>(
      cat, W8, xbuf, NN, 96, 192, 0, 0, 0, b8, 1.0f, 0, nullptr, 0);
  out_kernel<<<(BB * PRED * CC + 255) / 256, 256, 0, stream>>>(xbuf, (float*)d_out);
}